// EvidenceQualityLayer_63943473103359
// MI455X (gfx1250) — compile-verified
//
#include <hip/hip_runtime.h>

// EvidenceQualityLayer for MI455X (gfx1250), fp32 WMMA (V_WMMA_F32_16X16X4_F32)
// + gfx1250 async global->LDS staging (GLOBAL_LOAD_ASYNC_TO_LDS_B128, ASYNCcnt).
// Pipeline: 3x GEMM (QKV proj, fused head-split) -> fused evidence attention
// (flash-style, no materialized scores) -> 1x GEMM (output proj).
// Workspace requirement: 4 * 4096*1024 floats = 64 MB.

#define B_SZ 2
#define SEQ  2048
#define DM   1024
#define NH   16
#define HD   64
#define MROWS (B_SZ * SEQ)

typedef __attribute__((ext_vector_type(2))) float v2f;
typedef __attribute__((ext_vector_type(8))) float v8f;
typedef int v4i __attribute__((__vector_size__(16)));

__device__ __forceinline__ v8f v8zero() {
  v8f z;
#pragma unroll
  for (int i = 0; i < 8; ++i) z[i] = 0.0f;
  return z;
}

#define WMMA4(a, b, c) \
  __builtin_amdgcn_wmma_f32_16x16x4_f32(false, (a), false, (b), (short)0, (c), false, false)

// ---- gfx1250 async global->LDS copy (16B per lane), with safe fallback ----
#if __has_builtin(__builtin_amdgcn_global_load_async_to_lds_b128)
#define HAVE_ASYNC_LDS 1
#else
#define HAVE_ASYNC_LDS 0
#endif

__device__ __forceinline__ void async_copy16(const float* g, float* l) {
#if HAVE_ASYNC_LDS
  // Builtin signature (probe-confirmed): (v4i as1*, v4i as3*, imm, imm).
  // AS casts via uintptr_t: global generic addr == AS1 addr; LDS generic
  // addr low 32 bits == LDS offset (aperture lives in the high dword).
  __builtin_amdgcn_global_load_async_to_lds_b128(
      (__attribute__((address_space(1))) v4i*)(uintptr_t)g,
      (__attribute__((address_space(3))) v4i*)(uintptr_t)l, 0, 0);
#else
  *(float4*)l = *(const float4*)g;
#endif
}

__device__ __forceinline__ void async_wait_all() {
#if HAVE_ASYNC_LDS
#if __has_builtin(__builtin_amdgcn_s_wait_asynccnt)
  __builtin_amdgcn_s_wait_asynccnt(0);
#else
  asm volatile("s_wait_asynccnt 0x0" ::: "memory");
#endif
#endif
}

// ---------------- GEMM (NT): out = A[M,K] @ W[N,K]^T + bias ----------------
// 256 threads = 8 waves; 128x128 tile; K-chunk 32; LDS stride 36 (bank-conflict
// free for the 16-lane fragment reads, 16B aligned for the async b128 stores).
#define GT_M 128
#define GT_N 128
#define GT_K 32
#define GPAD 36

__global__ __launch_bounds__(256) void gemm_nt_wmma(
    const float* __restrict__ A, const float* __restrict__ W,
    const float* __restrict__ bias, float* __restrict__ out,
    int N, int K, int headsplit)
{
  __shared__ float As[GT_M * GPAD];
  __shared__ float Ws[GT_N * GPAD];

  const int tid  = threadIdx.x;
  const int lane = tid & 31;
  const int wid  = tid >> 5;
  const int bm = blockIdx.x * GT_M;
  const int bn = blockIdx.y * GT_N;
  const int wm = (wid >> 2) * 64;      // 0,64
  const int wn = (wid & 3) * 32;       // 0,32,64,96
  const int lrow  = lane & 15;         // M (A) / N (B) within 16
  const int khalf = (lane >> 4) << 1;  // K pair base: 0 or 2

  v8f acc[4][2];
#pragma unroll
  for (int i = 0; i < 4; ++i)
#pragma unroll
    for (int j = 0; j < 2; ++j) acc[i][j] = v8zero();

  const int ldr = tid >> 3;        // 0..31
  const int ldc = (tid & 7) << 2;  // 0..28

  for (int k0 = 0; k0 < K; k0 += GT_K) {
#pragma unroll
    for (int p = 0; p < 4; ++p) {
      int r = p * 32 + ldr;
      async_copy16(A + (size_t)(bm + r) * K + k0 + ldc, &As[r * GPAD + ldc]);
      async_copy16(W + (size_t)(bn + r) * K + k0 + ldc, &Ws[r * GPAD + ldc]);
    }
    async_wait_all();
    __syncthreads();

    if (k0 + GT_K < K) {  // gfx1250 global_prefetch_b8 for the next K tile
      __builtin_prefetch(A + (size_t)(bm + ldr) * K + k0 + GT_K, 0, 1);
      __builtin_prefetch(W + (size_t)(bn + ldr) * K + k0 + GT_K, 0, 1);
    }

#pragma unroll
    for (int kk = 0; kk < GT_K; kk += 4) {
      v2f af[4], wf[2];
#pragma unroll
      for (int i = 0; i < 4; ++i)
        af[i] = *(const v2f*)&As[(wm + i * 16 + lrow) * GPAD + kk + khalf];
#pragma unroll
      for (int j = 0; j < 2; ++j)
        wf[j] = *(const v2f*)&Ws[(wn + j * 16 + lrow) * GPAD + kk + khalf];
#pragma unroll
      for (int i = 0; i < 4; ++i)
#pragma unroll
        for (int j = 0; j < 2; ++j)
          acc[i][j] = WMMA4(af[i], wf[j], acc[i][j]);
    }
    __syncthreads();
  }

  // Epilogue: bias + optional [B,S,D] -> [B,H,S,Hd] head split.
  const int crow0 = (lane >> 4) << 3;  // 0 or 8
  const int ccol  = lane & 15;
#pragma unroll
  for (int i = 0; i < 4; ++i)
#pragma unroll
    for (int j = 0; j < 2; ++j)
#pragma unroll
      for (int v = 0; v < 8; ++v) {
        int m = bm + wm + i * 16 + crow0 + v;
        int n = bn + wn + j * 16 + ccol;
        float val = acc[i][j][v] + bias[n];
        if (headsplit) {
          int b = m >> 11, s = m & (SEQ - 1);
          int h = n >> 6,  hd = n & (HD - 1);
          out[((size_t)((b * NH + h) * SEQ + s) << 6) + hd] = val;
        } else {
          out[(size_t)m * N + n] = val;
        }
      }
}

// ---------------- Fused evidence attention ----------------
// One WG (8 waves) per (b,h, 64-query block). Streams 64-key blocks:
//   S = Q K^T / 8 ; E = exp(S)*scale + (1+bias) ; denom += rowsum(E) ;
//   O += E V  (unnormalized) ; finalize O/denom, uncertainty = S/denom.
#define APAD 68

__global__ __launch_bounds__(256) void evidence_attn_wmma(
    const float* __restrict__ Q, const float* __restrict__ K,
    const float* __restrict__ V, float* __restrict__ attn,
    float* __restrict__ unc,
    const float* __restrict__ ev_scale_p, const float* __restrict__ ev_bias_p)
{
  __shared__ float Ks[64 * APAD];   // key-major
  __shared__ float Vs[64 * APAD];   // key-major (async-staged)
  __shared__ float Es[64 * APAD];   // evidence tile (C-layout -> A-layout)
  __shared__ float dsum[64];

  const int tid  = threadIdx.x;
  const int lane = tid & 31;
  const int wid  = tid >> 5;
  const int bh = blockIdx.y;          // b*NH + h
  const int q0 = blockIdx.x * 64;
  const int wm = (wid >> 1) << 4;     // 0,16,32,48 (query rows)
  const int wn = (wid & 1) << 5;      // 0,32 (key / head-dim cols)
  const int lrow  = lane & 15;
  const int khalf = (lane >> 4) << 1;
  const int crow0 = (lane >> 4) << 3;
  const int ccol  = lane & 15;

  const float ev_scale = ev_scale_p[0];
  const float ev_c = 1.0f + ev_bias_p[0];
  const size_t head_base = (size_t)bh * SEQ * HD;

  // Preload this wave's Q fragments (its 16 query rows, full Hd=64).
  v2f qf[16];
  {
    const float* qrow = Q + head_base + (size_t)(q0 + wm + lrow) * HD;
#pragma unroll
    for (int t = 0; t < 16; ++t)
      qf[t] = *(const v2f*)(qrow + t * 4 + khalf);
  }

  v8f oacc[2];
  oacc[0] = v8zero(); oacc[1] = v8zero();
  float dacc = 0.0f;

  const int ldr16 = tid >> 4;        // 0..15
  const int ldc4  = (tid & 15) << 2; // 0..60

  for (int kb = 0; kb < SEQ / 64; ++kb) {
    __syncthreads();  // previous block's LDS reads complete
    const float* Kbase = K + head_base + (size_t)(kb * 64) * HD;
    const float* Vbase = V + head_base + (size_t)(kb * 64) * HD;
#pragma unroll
    for (int p = 0; p < 4; ++p) {
      int r = p * 16 + ldr16;  // key index in block
      async_copy16(Kbase + (size_t)r * HD + ldc4, &Ks[r * APAD + ldc4]);
      async_copy16(Vbase + (size_t)r * HD + ldc4, &Vs[r * APAD + ldc4]);
    }
    async_wait_all();
    __syncthreads();

    // Scores: 16(q) x 32(key) per wave, K-loop over Hd.
    v8f sacc[2];
    sacc[0] = v8zero(); sacc[1] = v8zero();
#pragma unroll
    for (int t = 0; t < 16; ++t) {
#pragma unroll
      for (int j = 0; j < 2; ++j) {
        v2f kf = *(const v2f*)&Ks[(wn + j * 16 + lrow) * APAD + t * 4 + khalf];
        sacc[j] = WMMA4(qf[t], kf, sacc[j]);
      }
    }

    // Evidence transform; stage to LDS (layout conversion C -> A).
#pragma unroll
    for (int j = 0; j < 2; ++j)
#pragma unroll
      for (int v = 0; v < 8; ++v) {
        float s = sacc[j][v] * 0.125f;  // 1/sqrt(Hd)
        Es[(wm + crow0 + v) * APAD + wn + j * 16 + ccol] =
            __expf(s) * ev_scale + ev_c;
      }
    __syncthreads();

    // Evidence row-sums (denominator).
    if (tid < 64) {
      float sum = 0.0f;
#pragma unroll 8
      for (int c = 0; c < 64; ++c) sum += Es[tid * APAD + c];
      dacc += sum;
    }

    // O += E @ V : 16(q) x 32(d) per wave, K-loop over 64 keys.
#pragma unroll
    for (int t = 0; t < 16; ++t) {
      const int kr = t * 4 + khalf;
      v2f ef = *(const v2f*)&Es[(wm + lrow) * APAD + kr];
#pragma unroll
      for (int j = 0; j < 2; ++j) {
        const int n = wn + j * 16 + lrow;
        v2f vf;
        vf[0] = Vs[kr * APAD + n];
        vf[1] = Vs[(kr + 1) * APAD + n];
        oacc[j] = WMMA4(ef, vf, oacc[j]);
      }
    }
  }

  if (tid < 64) {
    dsum[tid] = dacc;
    unc[(size_t)bh * SEQ + q0 + tid] = (float)SEQ / dacc;
  }
  __syncthreads();

  // Normalize and write attn in [B,S,D] layout (d = h*64 + hd).
  const int b = bh >> 4, h = bh & (NH - 1);
#pragma unroll
  for (int j = 0; j < 2; ++j)
#pragma unroll
    for (int v = 0; v < 8; ++v) {
      int r = wm + crow0 + v;
      int dcol = wn + j * 16 + ccol;
      attn[(size_t)(b * SEQ + q0 + r) * DM + h * HD + dcol] = oacc[j][v] / dsum[r];
    }
}

extern "C" void kernel_launch(void* const* d_in, const int* in_sizes, int n_in,
                              void* d_out, int out_size, void* d_ws, size_t ws_size,
                              hipStream_t stream) {
  (void)in_sizes; (void)n_in; (void)out_size; (void)ws_size;
  const float* x   = (const float*)d_in[0];
  const float* q_w = (const float*)d_in[1];
  const float* q_b = (const float*)d_in[2];
  const float* k_w = (const float*)d_in[3];
  const float* k_b = (const float*)d_in[4];
  const float* v_w = (const float*)d_in[5];
  const float* v_b = (const float*)d_in[6];
  const float* o_w = (const float*)d_in[7];
  const float* o_b = (const float*)d_in[8];
  const float* es  = (const float*)d_in[9];
  const float* eb  = (const float*)d_in[10];

  float* out = (float*)d_out;
  float* unc = out + (size_t)MROWS * DM;

  float* Qh = (float*)d_ws;                 // [B,H,S,Hd]
  float* Kh = Qh + (size_t)MROWS * DM;
  float* Vh = Kh + (size_t)MROWS * DM;
  float* At = Vh + (size_t)MROWS * DM;      // [B,S,D]

  dim3 tb(256);
  dim3 gg(MROWS / GT_M, DM / GT_N);
  gemm_nt_wmma<<<gg, tb, 0, stream>>>(x, q_w, q_b, Qh, DM, DM, 1);
  gemm_nt_wmma<<<gg, tb, 0, stream>>>(x, k_w, k_b, Kh, DM, DM, 1);
  gemm_nt_wmma<<<gg, tb, 0, stream>>>(x, v_w, v_b, Vh, DM, DM, 1);
  evidence_attn_wmma<<<dim3(SEQ / 64, B_SZ * NH), tb, 0, stream>>>(
      Qh, Kh, Vh, At, unc, es, eb);
  gemm_nt_wmma<<<gg, tb, 0, stream>>>(At, o_w, o_b, out, DM, DM, 0);
}